// BaseModel_63058709840114
// MI455X (gfx1250) — compile-verified
//
#include <hip/hip_runtime.h>

// MI455X (gfx1250) wave32 WMMA implementation of the tiny-MLP reference.
// One wave = one 16-row tile. All 4 matmuls use V_WMMA_F32_16X16X4_F32
// (K padded to multiples of 4, N padded to 16 with zero weight columns).
// Layernorm runs in A-layout: per-lane partial sums over 6 elements +
// one v_permlanex16_b32 cross-half combine (no LDS shuffles, no waits).
// D->A layout conversions go through a per-wave LDS tile (stride-20 pad).

typedef __attribute__((ext_vector_type(2))) float v2f;
typedef __attribute__((ext_vector_type(8))) float v8f;

#define LDPITCH 20          // floats per LDS row (16 + 4 pad, avoids bank conflicts)
#define WAVES_PER_BLOCK 8

// Exchange with the same lane position in the other 16-lane half (pure VALU).
static __device__ __forceinline__ float permx16(float v) {
  int i = __builtin_bit_cast(int, v);
  i = __builtin_amdgcn_permlanex16(i, i, 0x76543210u, 0xfedcba98u, false, false);
  return __builtin_bit_cast(float, i);
}

static __device__ __forceinline__ v8f wmma3(const v2f a[3], const v2f b[3], v8f acc) {
  acc = __builtin_amdgcn_wmma_f32_16x16x4_f32(false, a[0], false, b[0], (short)0, acc, false, false);
  acc = __builtin_amdgcn_wmma_f32_16x16x4_f32(false, a[1], false, b[1], (short)0, acc, false, false);
  acc = __builtin_amdgcn_wmma_f32_16x16x4_f32(false, a[2], false, b[2], (short)0, acc, false, false);
  return acc;
}

// Layernorm over d=12 in A layout: lane holds K = {4f+2*half, +1} for f=0..2
// (6 elements); the other 6 elements of the same row live in lane l^16.
static __device__ __forceinline__ void ln_a(v2f a[3], const v2f g[3], const v2f b[3]) {
  float s = (a[0].x + a[0].y) + (a[1].x + a[1].y) + (a[2].x + a[2].y);
  float q = a[0].x * a[0].x;
  q = fmaf(a[0].y, a[0].y, q);
  q = fmaf(a[1].x, a[1].x, q);
  q = fmaf(a[1].y, a[1].y, q);
  q = fmaf(a[2].x, a[2].x, q);
  q = fmaf(a[2].y, a[2].y, q);
  s += permx16(s);
  q += permx16(q);
  float mu  = s * (1.0f / 12.0f);
  float var = q * (1.0f / 12.0f) - mu * mu;
  float rs  = __builtin_amdgcn_rsqf(var + 1e-5f);
#pragma unroll
  for (int f = 0; f < 3; ++f) {
    a[f].x = (a[f].x - mu) * rs * g[f].x + b[f].x;
    a[f].y = (a[f].y - mu) * rs * g[f].y + b[f].y;
  }
}

// D layout (rows across VGPRs, cols across lanes) -> A layout (M=lane%16, K across
// VGPRs/halves) via per-wave LDS tile. Same-wave DS ops are processed in order by
// the LDS unit; the asm barriers only pin the compiler's ordering.
static __device__ __forceinline__ void xpose(float* lb, v8f d, int col, int half,
                                             v2f a[3]) {
#pragma unroll
  for (int i = 0; i < 8; ++i)
    lb[(i + 8 * half) * LDPITCH + col] = d[i];
  __asm__ volatile("" ::: "memory");
#pragma unroll
  for (int f = 0; f < 3; ++f)
    a[f] = *(const v2f*)(lb + col * LDPITCH + 4 * f + 2 * half);   // ds_load_b64
  __asm__ volatile("" ::: "memory");
}

// B-matrix fragment for a [kdim x 12] row-major weight, frag f covers K=4f..4f+3.
// Lane mapping mirrors the A 16x4 layout transposed: N = lane%16, K = 4f + 2*half + j.
static __device__ __forceinline__ v2f wfrag(const float* W, int kdim, int f, int half,
                                            int col) {
  int k0 = 4 * f + 2 * half;
  v2f r; r.x = 0.0f; r.y = 0.0f;
  if (col < 12) {
    if (k0     < kdim) r.x = W[(size_t)k0       * 12 + col];
    if (k0 + 1 < kdim) r.y = W[(size_t)(k0 + 1) * 12 + col];
  }
  return r;
}

// Per-K layernorm parameter fragment (d = 12, k0 always < 12).
static __device__ __forceinline__ v2f kfrag(const float* p, int f, int half) {
  int k0 = 4 * f + 2 * half;
  v2f r; r.x = p[k0]; r.y = p[k0 + 1];
  return r;
}

template <bool FULL>
static __device__ __forceinline__ void do_tile(
    int t, int nrows, const float* __restrict__ x, float* __restrict__ out, float* lb,
    int col, int half,
    const v2f* fE, const v2f* f1, const v2f* f2, const v2f* fH,
    const v2f* g1k, const v2f* b1k, const v2f* ghk, const v2f* bhk,
    float bEl, float b1l, float b2l, float bHl) {
  const int row = t * 16 + col;   // A-layout: M = lane%16

  // ---- embed A fragments straight from global (aligned float2, row stride 40B) ----
  v2f a[3];
  if (FULL) {
    const float* xr = x + (size_t)row * 10 + 2 * half;
    a[0] = *(const v2f*)(xr);
    a[1] = *(const v2f*)(xr + 4);
    a[2].x = 0.0f; a[2].y = 0.0f;
    if (half == 0) a[2] = *(const v2f*)(x + (size_t)row * 10 + 8);
  } else {
#pragma unroll
    for (int f = 0; f < 3; ++f) {
      int k0 = 4 * f + 2 * half;
      v2f av; av.x = 0.0f; av.y = 0.0f;
      if (k0 < 10 && row < nrows) av = *(const v2f*)(x + (size_t)row * 10 + k0);
      a[f] = av;
    }
  }

  // ---- embed ----
  v8f h = {};
  h = wmma3(a, fE, h);
#pragma unroll
  for (int i = 0; i < 8; ++i) h[i] += bEl;

  // ---- LN1 (A-domain) -> fc1 -> ReLU ----
  xpose(lb, h, col, half, a);
  ln_a(a, g1k, b1k);
  v8f tacc = {};
  tacc = wmma3(a, f1, tacc);
#pragma unroll
  for (int i = 0; i < 8; ++i) tacc[i] = fmaxf(tacc[i] + b1l, 0.0f);

  // ---- fc2 + residual ----
  xpose(lb, tacc, col, half, a);
  v8f m = {};
  m = wmma3(a, f2, m);
#pragma unroll
  for (int i = 0; i < 8; ++i) h[i] += m[i] + b2l;

  // ---- LNh (A-domain) -> head ----
  xpose(lb, h, col, half, a);
  ln_a(a, ghk, bhk);
  v8f o = {};
  o = wmma3(a, fH, o);
#pragma unroll
  for (int i = 0; i < 8; ++i) o[i] += bHl;

  // ---- store: D layout VGPR i -> (row = t*16 + i + 8*half, col), cols 0..11 ----
  if (col < 12) {
#pragma unroll
    for (int i = 0; i < 8; ++i) {
      int r = t * 16 + i + 8 * half;
      if (FULL || r < nrows) out[(size_t)r * 12 + col] = o[i];
    }
  }
}

__global__ __launch_bounds__(256) void mlp_residual_kernel(
    const float* __restrict__ x,
    const float* __restrict__ w_embed, const float* __restrict__ b_embed,
    const float* __restrict__ g_norm1, const float* __restrict__ b_norm1,
    const float* __restrict__ w_fc1,   const float* __restrict__ b_fc1,
    const float* __restrict__ w_fc2,   const float* __restrict__ b_fc2,
    const float* __restrict__ g_normh, const float* __restrict__ b_normh,
    const float* __restrict__ w_head,  const float* __restrict__ b_head,
    float* __restrict__ out, int nrows) {
  __shared__ float lds[WAVES_PER_BLOCK * 16 * LDPITCH];

  const int lane = threadIdx.x & 31;
  const int wv   = threadIdx.x >> 5;
  const int half = lane >> 4;
  const int col  = lane & 15;
  float* lb = &lds[wv * 16 * LDPITCH];

  // ---- persistent per-wave state: weight frags + per-lane params ----
  v2f fE[3], f1[3], f2[3], fH[3];
  v2f g1k[3], b1k[3], ghk[3], bhk[3];
#pragma unroll
  for (int f = 0; f < 3; ++f) {
    fE[f]  = wfrag(w_embed, 10, f, half, col);
    f1[f]  = wfrag(w_fc1,   12, f, half, col);
    f2[f]  = wfrag(w_fc2,   12, f, half, col);
    fH[f]  = wfrag(w_head,  12, f, half, col);
    g1k[f] = kfrag(g_norm1, f, half);
    b1k[f] = kfrag(b_norm1, f, half);
    ghk[f] = kfrag(g_normh, f, half);
    bhk[f] = kfrag(b_normh, f, half);
  }
  const bool cv = (col < 12);
  const float bEl = cv ? b_embed[col] : 0.0f;
  const float b1l = cv ? b_fc1[col]   : 0.0f;
  const float b2l = cv ? b_fc2[col]   : 0.0f;
  const float bHl = cv ? b_head[col]  : 0.0f;

  const int ntiles     = (nrows + 15) >> 4;
  const int totalWaves = (int)gridDim.x * WAVES_PER_BLOCK;
  const int tid        = (int)blockIdx.x * WAVES_PER_BLOCK + wv;

  for (int t = tid; t < ntiles; t += totalWaves) {
    if (t * 16 + 16 <= nrows) {
      do_tile<true >(t, nrows, x, out, lb, col, half, fE, f1, f2, fH,
                     g1k, b1k, ghk, bhk, bEl, b1l, b2l, bHl);
    } else {
      do_tile<false>(t, nrows, x, out, lb, col, half, fE, f1, f2, fH,
                     g1k, b1k, ghk, bhk, bEl, b1l, b2l, bHl);
    }
  }
}

extern "C" void kernel_launch(void* const* d_in, const int* in_sizes, int n_in,
                              void* d_out, int out_size, void* d_ws, size_t ws_size,
                              hipStream_t stream) {
  (void)n_in; (void)out_size; (void)d_ws; (void)ws_size;
  const float* x       = (const float*)d_in[0];
  const float* w_embed = (const float*)d_in[1];
  const float* b_embed = (const float*)d_in[2];
  const float* g_norm1 = (const float*)d_in[3];
  const float* b_norm1 = (const float*)d_in[4];
  const float* w_fc1   = (const float*)d_in[5];
  const float* b_fc1   = (const float*)d_in[6];
  const float* w_fc2   = (const float*)d_in[7];
  const float* b_fc2   = (const float*)d_in[8];
  const float* g_normh = (const float*)d_in[9];
  const float* b_normh = (const float*)d_in[10];
  const float* w_head  = (const float*)d_in[11];
  const float* b_head  = (const float*)d_in[12];
  float* out = (float*)d_out;

  const int nrows  = in_sizes[0] / 10;
  const int ntiles = (nrows + 15) / 16;
  int blocks = (ntiles + WAVES_PER_BLOCK - 1) / WAVES_PER_BLOCK;
  if (blocks > 4096) blocks = 4096;
  if (blocks < 1) blocks = 1;

  mlp_residual_kernel<<<blocks, 256, 0, stream>>>(
      x, w_embed, b_embed, g_norm1, b_norm1, w_fc1, b_fc1,
      w_fc2, b_fc2, g_normh, b_normh, w_head, b_head, out, nrows);
}